// PseudoGroupContrast_pre_65506841198979
// MI455X (gfx1250) — compile-verified
//
#include <hip/hip_runtime.h>
#include <hip/hip_bf16.h>

typedef float v2f __attribute__((ext_vector_type(2)));
typedef float v8f __attribute__((ext_vector_type(8)));

#define CLASS_NUM  7
#define QUEUE_SIZE 168
#define PROJ_DIM   128
#define NCOLS      (CLASS_NUM * QUEUE_SIZE)   // 1176
#define NTILES     ((NCOLS + 15) / 16)        // 74
#define TEMP       0.5f
#define EPS        1e-8f
#define BATCHSZ    16384

// ---------------------------------------------------------------------------
// Kernel 1: per-row norms + l_pos.  One wave32 per row (128 floats = 4/lane).
// ---------------------------------------------------------------------------
__global__ __launch_bounds__(256) void pgc_norms_kernel(
    const float* __restrict__ act, const float* __restrict__ ema,
    float* __restrict__ inv_scale, float* __restrict__ l_pos, int nrows)
{
    int wave = (blockIdx.x * blockDim.x + threadIdx.x) >> 5;
    int lane = threadIdx.x & 31;
    if (wave >= nrows) return;

    const float4* a4 = (const float4*)(act + (size_t)wave * PROJ_DIM);
    const float4* e4 = (const float4*)(ema + (size_t)wave * PROJ_DIM);
    float4 a = a4[lane];
    float4 e = e4[lane];

    float saa = a.x*a.x + a.y*a.y + a.z*a.z + a.w*a.w;
    float see = e.x*e.x + e.y*e.y + e.z*e.z + e.w*e.w;
    float sae = a.x*e.x + a.y*e.y + a.z*e.z + a.w*e.w;

    #pragma unroll
    for (int m = 16; m >= 1; m >>= 1) {
        saa += __shfl_xor(saa, m, 32);
        see += __shfl_xor(see, m, 32);
        sae += __shfl_xor(sae, m, 32);
    }
    if (lane == 0) {
        float ra = __frsqrt_rn(saa);
        float re = __frsqrt_rn(see);
        inv_scale[wave] = ra * (1.0f / TEMP);  // applied to <act, q> before exp
        l_pos[wave]     = sae * ra * re;
    }
}

// ---------------------------------------------------------------------------
// Kernel 2: fused GEMM (f32 WMMA) + exp + per-row total / class-segment
// gather + log terms.  One workgroup per 16-row block, 4 waves.
// ---------------------------------------------------------------------------
__global__ __launch_bounds__(128) void pgc_main_kernel(
    const float* __restrict__ act, const float* __restrict__ queue,
    const int* __restrict__ label, const float* __restrict__ inv_scale,
    const float* __restrict__ l_pos, float* __restrict__ partials, int nrows)
{
    __shared__ float a_lds[16 * 132];            // padded stride: no bank conflicts
    __shared__ float pos_lds[16 * QUEUE_SIZE];   // class-segment exponentials
    __shared__ float tot_lds[16];
    __shared__ float denom_lds[16];
    __shared__ float red[128];

    const int tid    = threadIdx.x;
    const int waveid = tid >> 5;
    const int lane   = tid & 31;
    const int rowbase = blockIdx.x * 16;

    if (tid < 16) tot_lds[tid] = 0.0f;

    // ---- stage A tile (16 x 128 f32) into LDS, float4 granularity ----
    const float* Ag = act + (size_t)rowbase * PROJ_DIM;
    for (int i = tid; i < 16 * 32; i += 128) {
        int r  = i >> 5;      // 32 float4 per row
        int c4 = i & 31;
        float4 v = ((const float4*)(Ag + r * PROJ_DIM))[c4];
        float* dst = &a_lds[r * 132 + c4 * 4];
        dst[0] = v.x; dst[1] = v.y; dst[2] = v.z; dst[3] = v.w;
    }
    __syncthreads();

    const int mrow = lane & 15;     // A-fragment row for this lane
    const int hi   = lane >> 4;     // K sub-offset selector

    // C-fragment rows for this lane: r = v + 8*hi, v = 0..7
    float invs[8]; int lab[8];
    #pragma unroll
    for (int v = 0; v < 8; ++v) {
        int r = v + 8 * hi;
        invs[v] = inv_scale[rowbase + r];
        lab[v]  = label[rowbase + r];
    }

    float totpart[8] = {0, 0, 0, 0, 0, 0, 0, 0};

    const float* Al = &a_lds[mrow * 132];

    for (int t = waveid; t < NTILES; t += 4) {
        int colbase = t * 16;
        int mycol   = colbase + (lane & 15);
        int safecol = mycol < NCOLS ? mycol : NCOLS - 1;
        const float* Bg = queue + (size_t)safecol * PROJ_DIM;

        v8f c = {};
        #pragma unroll
        for (int s = 0; s < 32; ++s) {
            int k = 4 * s + 2 * hi;
            v2f a = *(const v2f*)(Al + k);   // LDS b64 read
            v2f b = *(const v2f*)(Bg + k);   // global b64 read (queue row-major = B k-major)
            c = __builtin_amdgcn_wmma_f32_16x16x4_f32(
                    false, a, false, b, (short)0, c, false, false);
        }

        bool colvalid = mycol < NCOLS;
        int  cls = mycol / QUEUE_SIZE;
        int  cq  = mycol - cls * QUEUE_SIZE;
        #pragma unroll
        for (int v = 0; v < 8; ++v) {
            if (colvalid) {
                float e = __expf(c[v] * invs[v]);
                totpart[v] += e;
                int r = v + 8 * hi;
                if (cls == lab[v]) pos_lds[r * QUEUE_SIZE + cq] = e;
            }
        }
    }

    // ---- reduce per-row totals across lanes/waves via LDS float atomics ----
    #pragma unroll
    for (int v = 0; v < 8; ++v)
        atomicAdd(&tot_lds[v + 8 * hi], totpart[v]);
    __syncthreads();

    if (tid < 16)
        denom_lds[tid] = l_pos[rowbase + tid] + tot_lds[tid];
    __syncthreads();

    // ---- loss terms ----
    float acc = 0.0f;
    for (int idx = tid; idx < 16 * QUEUE_SIZE; idx += 128) {
        int r = idx / QUEUE_SIZE;
        acc += -__logf(pos_lds[idx] / denom_lds[r] + EPS);
    }
    if (tid < 16) {
        float lp = l_pos[rowbase + tid];
        acc += -__logf(lp / denom_lds[tid] + EPS);
    }

    red[tid] = acc;
    __syncthreads();
    #pragma unroll
    for (int s = 64; s > 0; s >>= 1) {
        if (tid < s) red[tid] += red[tid + s];
        __syncthreads();
    }
    if (tid == 0)
        partials[blockIdx.x] = red[0] * (1.0f / ((QUEUE_SIZE + 1) * (float)BATCHSZ));
}

// ---------------------------------------------------------------------------
// Kernel 3: final reduction of per-workgroup partials -> scalar loss.
// ---------------------------------------------------------------------------
__global__ __launch_bounds__(256) void pgc_reduce_kernel(
    const float* __restrict__ partials, float* __restrict__ out, int n)
{
    __shared__ float red[256];
    float a = 0.0f;
    for (int i = threadIdx.x; i < n; i += 256) a += partials[i];
    red[threadIdx.x] = a;
    __syncthreads();
    #pragma unroll
    for (int s = 128; s > 0; s >>= 1) {
        if (threadIdx.x < s) red[threadIdx.x] += red[threadIdx.x + s];
        __syncthreads();
    }
    if (threadIdx.x == 0) out[0] = red[0];
}

// ---------------------------------------------------------------------------
extern "C" void kernel_launch(void* const* d_in, const int* in_sizes, int n_in,
                              void* d_out, int out_size, void* d_ws, size_t ws_size,
                              hipStream_t stream) {
    const float* act   = (const float*)d_in[0];
    const float* ema   = (const float*)d_in[1];
    const int*   label = (const int*)d_in[2];
    const float* queue = (const float*)d_in[3];

    const int nrows = in_sizes[2];          // 16384
    const int nblk  = nrows / 16;           // 1024 row-blocks

    float* inv_scale = (float*)d_ws;
    float* l_pos     = inv_scale + nrows;
    float* partials  = l_pos + nrows;

    // 8 waves (rows) per 256-thread block
    pgc_norms_kernel<<<(nrows + 7) / 8, 256, 0, stream>>>(act, ema, inv_scale, l_pos, nrows);
    pgc_main_kernel<<<nblk, 128, 0, stream>>>(act, queue, label, inv_scale, l_pos, partials, nrows);
    pgc_reduce_kernel<<<1, 256, 0, stream>>>(partials, (float*)d_out, nblk);
}